// GNN_40029095199405
// MI455X (gfx1250) — compile-verified
//
#include <hip/hip_runtime.h>
#include <hip/hip_bf16.h>

#define DIM 128

typedef __attribute__((ext_vector_type(2))) float v2f;
typedef __attribute__((ext_vector_type(8))) float v8f;

// ---------------------------------------------------------------- utilities
__global__ __launch_bounds__(256) void zero_f32(float* __restrict__ p, long long n) {
  long long i = (long long)blockIdx.x * blockDim.x + threadIdx.x;
  long long stride = (long long)gridDim.x * blockDim.x;
  for (; i < n; i += stride) p[i] = 0.0f;
}

__global__ __launch_bounds__(256) void degree_kernel(const int* __restrict__ dst,
                                                     float* __restrict__ deg, int n_edges) {
  int e = blockIdx.x * blockDim.x + threadIdx.x;
  if (e < n_edges) atomicAdd(&deg[dst[e]], 1.0f);
}

__global__ __launch_bounds__(256) void invdeg_kernel(const float* __restrict__ deg,
                                                     float* __restrict__ invd, int n) {
  int i = blockIdx.x * blockDim.x + threadIdx.x;
  if (i < n) invd[i] = 1.0f / fmaxf(deg[i], 1.0f);
}

// One wave32 per edge: lane l moves feat[src][4l..4l+3] -> atomic add into agg[dst].
// The 512B row read is fully coalesced; atomics land in L2 (all 51.2MB of agg is L2-resident).
__global__ __launch_bounds__(256) void scatter_kernel(const float* __restrict__ feat,
                                                      const int* __restrict__ src,
                                                      const int* __restrict__ dst,
                                                      float* __restrict__ agg, int n_edges) {
  int e = blockIdx.x * 8 + (threadIdx.x >> 5);
  int lane = threadIdx.x & 31;
  if (e >= n_edges) return;
  int s = src[e];
  int d = dst[e];
  const float4 v = *(const float4*)(feat + (size_t)s * DIM + lane * 4);
  float* o = agg + (size_t)d * DIM + lane * 4;
  atomicAdd(o + 0, v.x);
  atomicAdd(o + 1, v.y);
  atomicAdd(o + 2, v.z);
  atomicAdd(o + 3, v.w);
}

// ---------------------------------------------------------------- fused SAGE GEMM
// out[m, :] = (agg[m,:] * invd[m]) @ Wl + bias + h[m,:] @ Wr   (+ optional ReLU)
// blockDim = 128 (4 waves); each wave owns a 16-row x 128-col output strip.
// Uses V_WMMA_F32_16X16X4_F32 (fp32 WMMA), weights staged in dynamic LDS (128.5KB, CDNA5 WGP
// has 320KB).
__global__ __launch_bounds__(128)
void sage_gemm(const float* __restrict__ agg, const float* __restrict__ invd,
               const float* __restrict__ h,
               const float* __restrict__ Wl, const float* __restrict__ Wr,
               const float* __restrict__ bias,
               float* __restrict__ out, int n_nodes, int do_relu)
{
  extern __shared__ float smem[];
  float* sWl = smem;                  // DIM*DIM
  float* sWr = smem + DIM * DIM;      // DIM*DIM
  float* sb  = smem + 2 * DIM * DIM;  // DIM

  int tid = threadIdx.x;
  for (int i = tid * 4; i < DIM * DIM; i += 128 * 4) {
    *(float4*)(sWl + i) = *(const float4*)(Wl + i);
    *(float4*)(sWr + i) = *(const float4*)(Wr + i);
  }
  if (tid < DIM) sb[tid] = bias[tid];
  __syncthreads();

  int wave = tid >> 5;
  int lane = tid & 31;
  int rowBase = (blockIdx.x * 4 + wave) * 16;
  if (rowBase >= n_nodes) return;

  // ISA A-matrix 16x4 f32 layout: lanes 0-15 hold M=lane with {K0,K1}, lanes 16-31 hold {K2,K3}
  int laneM = lane & 15;
  int kOff  = (lane >> 4) << 1;   // 0 or 2
  int row   = rowBase + laneM;
  float id  = invd[row];
  const float* aggRow = agg + (size_t)row * DIM;
  const float* hRow   = h   + (size_t)row * DIM;

  v8f acc[8] = {};  // 8 N-tiles of 16 cols; shared accumulator for both weight terms

#pragma unroll 2
  for (int k0 = 0; k0 < DIM; k0 += 4) {
    int k = k0 + kOff;
    v2f aA, aH;
    aA[0] = aggRow[k]     * id;
    aA[1] = aggRow[k + 1] * id;
    aH[0] = hRow[k];
    aH[1] = hRow[k + 1];
#pragma unroll
    for (int t = 0; t < 8; ++t) {
      int n = t * 16 + laneM;
      // B-matrix 4x16 f32 layout mirrors A: VGPR0 = rows {K0 | K2}, VGPR1 = rows {K1 | K3}
      v2f bl, br;
      bl[0] = sWl[k * DIM + n];
      bl[1] = sWl[(k + 1) * DIM + n];
      br[0] = sWr[k * DIM + n];
      br[1] = sWr[(k + 1) * DIM + n];
      acc[t] = __builtin_amdgcn_wmma_f32_16x16x4_f32(
          false, aA, false, bl, (short)0, acc[t], false, false);
      acc[t] = __builtin_amdgcn_wmma_f32_16x16x4_f32(
          false, aH, false, br, (short)0, acc[t], false, false);
    }
  }

  // C/D layout: VGPR r, lanes 0-15 -> (M=r, N=lane); lanes 16-31 -> (M=r+8, N=lane-16)
  int mOff = (lane >> 4) << 3;  // 0 or 8
#pragma unroll
  for (int t = 0; t < 8; ++t) {
    int n = t * 16 + laneM;
    float bv = sb[n];
#pragma unroll
    for (int r = 0; r < 8; ++r) {
      float v = acc[t][r] + bv;
      if (do_relu) v = fmaxf(v, 0.0f);
      out[(size_t)(rowBase + mOff + r) * DIM + n] = v;
    }
  }
}

// ---------------------------------------------------------------- launcher
extern "C" void kernel_launch(void* const* d_in, const int* in_sizes, int n_in,
                              void* d_out, int out_size, void* d_ws, size_t ws_size,
                              hipStream_t stream) {
  const float* x   = (const float*)d_in[0];
  const int*   ei  = (const int*)d_in[1];
  const float* W1l = (const float*)d_in[2];
  const float* b1  = (const float*)d_in[3];
  const float* W1r = (const float*)d_in[4];
  const float* W2l = (const float*)d_in[5];
  const float* b2  = (const float*)d_in[6];
  const float* W2r = (const float*)d_in[7];
  float* out = (float*)d_out;

  const int nNodes = in_sizes[0] / DIM;       // 100000
  const int nEdges = in_sizes[1] / 2;         // 1600000
  const int* src = ei;
  const int* dst = ei + nEdges;

  // workspace layout
  float* agg  = (float*)d_ws;                       // nNodes*DIM
  float* h1   = agg + (size_t)nNodes * DIM;         // nNodes*DIM
  float* deg  = h1  + (size_t)nNodes * DIM;         // nNodes
  float* invd = deg + nNodes;                       // nNodes

  const long long featN = (long long)nNodes * DIM;
  const int rowTiles   = (nNodes + 15) / 16;
  const int gemmBlocks = (rowTiles + 3) / 4;
  const size_t ldsBytes = (size_t)(2 * DIM * DIM + DIM) * sizeof(float);

  // degree (shared by both layers)
  zero_f32<<<256, 256, 0, stream>>>(deg, nNodes);
  degree_kernel<<<(nEdges + 255) / 256, 256, 0, stream>>>(dst, deg, nEdges);
  invdeg_kernel<<<(nNodes + 255) / 256, 256, 0, stream>>>(deg, invd, nNodes);

  // layer 1
  zero_f32<<<2048, 256, 0, stream>>>(agg, featN);
  scatter_kernel<<<(nEdges + 7) / 8, 256, 0, stream>>>(x, src, dst, agg, nEdges);
  sage_gemm<<<gemmBlocks, 128, ldsBytes, stream>>>(agg, invd, x, W1l, W1r, b1, h1, nNodes, 1);

  // layer 2
  zero_f32<<<2048, 256, 0, stream>>>(agg, featN);
  scatter_kernel<<<(nEdges + 7) / 8, 256, 0, stream>>>(h1, src, dst, agg, nEdges);
  sage_gemm<<<gemmBlocks, 128, ldsBytes, stream>>>(agg, invd, h1, W2l, W2r, b2, out, nNodes, 0);
}